// DHASPILevelLoss_43825846288900
// MI455X (gfx1250) — compile-verified
//
#include <hip/hip_runtime.h>

// ---------------- problem constants ----------------
#define N_SIG      512                 // B*C signals per tensor
#define T_LEN      192000              // samples per signal
#define SUB        960                 // gcd(FRAME, SHIFT)
#define NSUB       200                 // T_LEN / SUB
#define SUBS_TILE  8                   // sub-blocks per streamed tile
#define TILE_F     (SUB * SUBS_TILE)   // 7680 floats = 30 KB
#define TILE_V4    (TILE_F / 4)        // 1920 float4
#define NTILES     (NSUB / SUBS_TILE)  // 25
#define NFRM       64                  // (T - FRAME)/SHIFT + 1
#define BLOCK_THR  128                 // 4 waves (wave32)
#define WAVES      4
#define V4_PER_THR (TILE_V4 / BLOCK_THR) // 15 async b128 issues per thread per tile

// ---------------- CDNA5 async global->LDS path ----------------
#if defined(__AMDGCN__) && __has_builtin(__builtin_amdgcn_global_load_async_to_lds_b128)
  #define HAVE_ASYNC 1
#else
  #define HAVE_ASYNC 0
#endif

#if defined(__AMDGCN__) && __has_builtin(__builtin_amdgcn_s_wait_asynccnt)
  #define WAIT_ASYNC(n) __builtin_amdgcn_s_wait_asynccnt(n)
#elif HAVE_ASYNC
  #define WAIT_ASYNC(n) asm volatile("s_wait_asynccnt %0" ::"n"(n) : "memory")
#else
  #define WAIT_ASYNC(n) do {} while (0)
#endif

typedef int v4i_t __attribute__((ext_vector_type(4)));
#if HAVE_ASYNC
typedef __attribute__((address_space(1))) v4i_t g_v4i;   // global (AS1) int4
typedef __attribute__((address_space(3))) v4i_t l_v4i;   // LDS (AS3) int4
#endif

// Issue one 30KB tile: 15 x b128 per thread, uniform per wave so ASYNCcnt
// bookkeeping is exact (15 outstanding per in-flight tile per wave).
__device__ __forceinline__ void issue_tile(const float* __restrict__ g,
                                           float* __restrict__ l, int tid) {
#if HAVE_ASYNC
  #pragma unroll
  for (int k = 0; k < V4_PER_THR; ++k) {
    const int i = tid + k * BLOCK_THR;
    __builtin_amdgcn_global_load_async_to_lds_b128(
        (g_v4i*)(g + (size_t)i * 4),
        (l_v4i*)(l + i * 4),
        0, 0);
  }
#else
  #pragma unroll
  for (int k = 0; k < V4_PER_THR; ++k) {
    const int i = tid + k * BLOCK_THR;
    float4 v = reinterpret_cast<const float4*>(g)[i];
    reinterpret_cast<float4*>(l)[i] = v;
  }
#endif
}

// ---------------- kernel 1: per-signal gated loudness ----------------
__global__ __launch_bounds__(BLOCK_THR)
void dhasp_loudness_kernel(const float* __restrict__ x,
                           const float* __restrict__ y,
                           float* __restrict__ lufs) {
  __shared__ float buf[2][TILE_F];   // 61440 B double buffer
  __shared__ float subsum[NSUB];     // 800 B sub-block sum of squares
  __shared__ float zbuf[NFRM];
  __shared__ float elbuf[NFRM];

  const int tid = threadIdx.x;
  const int sig = blockIdx.x;                    // 0..1023 (x then y)
  const float* g = (sig < N_SIG) ? (x + (size_t)sig * T_LEN)
                                 : (y + (size_t)(sig - N_SIG) * T_LEN);

  const int wid  = tid >> 5;
  const int lane = tid & 31;

  // prime the double buffer
  issue_tile(g, buf[0], tid);
  issue_tile(g + TILE_F, buf[1], tid);

  for (int t = 0; t < NTILES; ++t) {
    // wait for tile t (its 15 per-wave loads), tile t+1 may stay in flight
    if (t + 1 < NTILES) { WAIT_ASYNC(V4_PER_THR); } else { WAIT_ASYNC(0); }
    __syncthreads();

    const float* lb = buf[t & 1];
    #pragma unroll
    for (int s2 = 0; s2 < 2; ++s2) {
      const int lsub = wid + s2 * WAVES;         // 0..7: sub-block within tile
      const float* p = lb + lsub * SUB;
      float acc = 0.0f;
      #pragma unroll
      for (int k = 0; k < SUB / 32; ++k) {       // 30 conflict-free LDS reads
        float v = p[k * 32 + lane];
        acc = fmaf(v, v, acc);
      }
      #pragma unroll
      for (int off = 16; off > 0; off >>= 1)     // wave32 reduction
        acc += __shfl_xor(acc, off, 32);
      if (lane == 0) subsum[t * SUBS_TILE + lsub] = acc;
    }
    __syncthreads();                              // buffer (t&1) now reusable

    if (t + 2 < NTILES) issue_tile(g + (size_t)(t + 2) * TILE_F, buf[t & 1], tid);
  }

  // ---- frame energies + gating (tiny: 64 frames) ----
  if (tid < NFRM) {
    float s = 0.0f;
    #pragma unroll
    for (int k = 0; k < 10; ++k) s += subsum[3 * tid + k];
    const float z = s * (1.0f / 9600.0f);        // mean square over FRAME
    zbuf[tid] = z;
    elbuf[tid] = -0.691f + 10.0f * log10f(z + 1e-8f);
  }
  __syncthreads();

  if (tid == 0) {
    float num = 0.0f, den = 0.0f;
    for (int f = 0; f < NFRM; ++f)
      if (elbuf[f] > -70.0f) { num += zbuf[f]; den += 1.0f; }
    const float z_ave_a = num / (den + 1e-8f);
    const float gamma_r = -0.691f + 10.0f * log10f(z_ave_a + 1e-8f) - 10.0f;

    float num2 = 0.0f, den2 = 0.0f;
    for (int f = 0; f < NFRM; ++f)
      if ((elbuf[f] > -70.0f) && (elbuf[f] > gamma_r)) { num2 += zbuf[f]; den2 += 1.0f; }
    const float z_ave_ar = num2 / (den2 + 1e-8f);

    lufs[sig] = -0.691f + 10.0f * log10f(z_ave_ar + 1e-8f);
  }
}

// ---------------- kernel 2: hinge + scale + sum ----------------
__global__ __launch_bounds__(N_SIG)
void dhasp_loss_kernel(const float* __restrict__ lufs, float* __restrict__ out) {
  __shared__ float red[N_SIG];
  const int i = threadIdx.x;
  const float d = lufs[N_SIG + i] - lufs[i];     // loud_y - loud_x
  red[i] = d > 0.0f ? d : 0.0f;
  __syncthreads();
  for (int s = N_SIG / 2; s > 0; s >>= 1) {
    if (i < s) red[i] += red[i + s];
    __syncthreads();
  }
  if (i == 0) out[0] = 1e-4f * red[0];
}

// ---------------- launch ----------------
extern "C" void kernel_launch(void* const* d_in, const int* in_sizes, int n_in,
                              void* d_out, int out_size, void* d_ws, size_t ws_size,
                              hipStream_t stream) {
  (void)in_sizes; (void)n_in; (void)out_size; (void)ws_size;
  const float* x = (const float*)d_in[0];
  const float* y = (const float*)d_in[1];
  float* lufs = (float*)d_ws;                    // 1024 floats scratch

  dhasp_loudness_kernel<<<dim3(2 * N_SIG), dim3(BLOCK_THR), 0, stream>>>(x, y, lufs);
  dhasp_loss_kernel<<<dim3(1), dim3(N_SIG), 0, stream>>>(lufs, (float*)d_out);
}